// BiLSTMCRF_87806311399529
// MI455X (gfx1250) — compile-verified
//
#include <hip/hip_runtime.h>

typedef __attribute__((ext_vector_type(2))) float v2f;
typedef __attribute__((ext_vector_type(8))) float v8f;
typedef __attribute__((ext_vector_type(4))) unsigned int v4u;
typedef __attribute__((ext_vector_type(8))) int v8i;
typedef __attribute__((ext_vector_type(4))) int v4i;

#define Bn     64
#define Tn     512
#define En     128
#define Hn     128
#define G4H    512           // 4*H
#define MTOT   (Bn*Tn)       // 32768
#define NTAGS  8

// scan kernel dynamic LDS: w_hh^T [128][513] + h^T [128][17] + xg stage [16][512]
#define SCAN_LDS_FLOATS (128*513 + 128*17 + 16*512)
#define SCAN_LDS_BYTES  (SCAN_LDS_FLOATS*4)

__device__ __forceinline__ float sigmf(float x){ return 1.0f/(1.0f + __expf(-x)); }
__device__ __forceinline__ v8f v8zero(){ v8f z = {0.f,0.f,0.f,0.f,0.f,0.f,0.f,0.f}; return z; }

// ---------------------------------------------------------------- embedding
__global__ void embed_k(const int* __restrict__ sent, const float* __restrict__ emb,
                        float* __restrict__ x0){
  int idx = blockIdx.x*blockDim.x + threadIdx.x;        // < B*T*E
  int bt  = idx >> 7;
  int e   = idx & 127;
  x0[idx] = emb[(size_t)sent[bt]*En + e];
}

// ------------------------------------------------- xg = X @ Wih^T + bih+bhh
// X: (MTOT,K) row-major, Wih: (512,K) row-major, XG: (MTOT,512)
// block = 256 (8 waves). block m-tile = 16 rows; wave w -> 64 N columns.
__global__ void gates_gemm_k(const float* __restrict__ X, const float* __restrict__ Wih,
                             const float* __restrict__ bih, const float* __restrict__ bhh,
                             float* __restrict__ XG, int K){
  __shared__ float ax[16*257];                          // A tile, padded stride K+1
  const int tid = threadIdx.x;
  const int m0  = blockIdx.x * 16;
  for (int idx = tid; idx < 16*K; idx += 256){
    int r = idx / K, c = idx - r*K;
    ax[r*(K+1) + c] = X[(size_t)(m0+r)*K + c];
  }
  __syncthreads();

  const int w    = tid >> 5;
  const int lane = tid & 31;
  const int lm   = lane & 15;
  const int hi   = lane >> 4;                           // half-wave
  const int koff = hi*2;
  const int n0   = w*64;

  v8f acc[4];
  #pragma unroll
  for (int s=0;s<4;++s) acc[s] = v8zero();

  for (int k = 0; k < K; k += 4){
    v2f a;                                              // A frag: 16x4, row=lm
    a[0] = ax[lm*(K+1) + k     + koff];
    a[1] = ax[lm*(K+1) + k + 1 + koff];
    #pragma unroll
    for (int s=0;s<4;++s){
      const int n = n0 + s*16 + lm;                     // B frag: col = n
      v2f bb;
      bb[0] = Wih[(size_t)n*K + k     + koff];
      bb[1] = Wih[(size_t)n*K + k + 1 + koff];
      acc[s] = __builtin_amdgcn_wmma_f32_16x16x4_f32(
                 false, a, false, bb, (short)0, acc[s], false, false);
    }
  }
  #pragma unroll
  for (int s=0;s<4;++s){
    const int n = n0 + s*16 + lm;
    const float bv = bih[n] + bhh[n];
    #pragma unroll
    for (int r=0;r<8;++r){
      const int row = m0 + r + hi*8;
      XG[(size_t)row*G4H + n] = acc[s][r] + bv;
    }
  }
}

// ---------------------------------------------- bidirectional LSTM scan
// grid (B/16, 2): x = batch group (16 rows), y = direction (0 fwd, 1 bwd)
// block 256 = 8 waves; wave j owns h/c tile (16 batch x 16 H-cols) + 4 gate tiles.
// Per step, the next step's 16x512 xg tile is prefetched into LDS by the
// Tensor Data Mover (tensor_load_to_lds, TENSORcnt) while WMMA runs.
__global__ void lstm_scan_k(const float* __restrict__ xg_f, const float* __restrict__ xg_b,
                            const float* __restrict__ whh_f, const float* __restrict__ whh_b,
                            float* __restrict__ Y){
  extern __shared__ float smem[];
  float* wlds  = smem;                                  // [k 0..127][n 0..511] stride 513
  float* hT    = smem + 128*513;                        // [k 0..127][m 0..15] stride 17
  float* stage = hT   + 128*17;                         // [16][512] xg tile for current step

  const int dir = blockIdx.y;
  const float* __restrict__ XG  = dir ? xg_b  : xg_f;
  const float* __restrict__ Whh = dir ? whh_b : whh_f;

  const int tid = threadIdx.x;
  const int b0  = blockIdx.x * 16;

  // preload w_hh transposed (coalesced global read, stride-513 LDS write)
  for (int idx = tid; idx < G4H*Hn; idx += 256){
    int n = idx >> 7, k = idx & 127;
    wlds[k*513 + n] = Whh[idx];
  }
  for (int idx = tid; idx < Hn*17; idx += 256) hT[idx] = 0.f;

  const unsigned stage_lds = (unsigned)((size_t)(stage) & 0xFFFFFFFFu); // LDS byte offset

  const int w    = tid >> 5;
  const int lane = tid & 31;
  const int lm   = lane & 15;
  const int hi   = lane >> 4;
  const int koff = hi*2;
  const int nh   = w*16;                                // H-column base of this wave
  const int outOff = dir*Hn;

  // TDM descriptor issue: 2-D tile, 16 rows x 512 f32, row stride T*512 elements
  auto issue_tdm = [&](int tt){
    unsigned long long ga =
        (unsigned long long)(const void*)(XG + (((size_t)b0*Tn + tt)*G4H));
    v4u g0;
    g0[0] = 1u;                                         // count=1 valid descriptor
    g0[1] = stage_lds;                                  // lds_addr
    g0[2] = (unsigned)ga;                               // global_addr[31:0]
    g0[3] = (unsigned)((ga >> 32) & 0x1FFFFFFu) | 0x80000000u; // addr[56:32] | type=2
    v8i g1;
    g1[0] = 0x00020000;                                 // data_size=2 (4B), no mask
    g1[1] = (int)(512u << 16);                          // tensor_dim0 = 512 (bits 79:48)
    g1[2] = (int)(16u  << 16);                          // tensor_dim1 = 16  (bits 111:80)
    g1[3] = (int)(512u << 16);                          // tile_dim0 = 512   (bits 127:112)
    g1[4] = 16;                                         // tile_dim1 = 16
    g1[5] = Tn*G4H;                                     // tensor_dim0_stride = 262144 elems
    g1[6] = 0;
    g1[7] = 0;
    v4i gz4 = {0,0,0,0};
    v8i gz8 = {0,0,0,0,0,0,0,0};
    __builtin_amdgcn_tensor_load_to_lds(g0, g1, gz4, gz4, gz8, 0);
  };

  float c[8];
  #pragma unroll
  for (int r=0;r<8;++r) c[r] = 0.f;

  if (tid < 32){                                        // wave 0 drives the TDM
    issue_tdm(dir ? (Tn-1) : 0);
    __builtin_amdgcn_s_wait_tensorcnt(0);
  }
  __syncthreads();                                      // wlds/hT init + first tile ready

  for (int s = 0; s < Tn; ++s){
    const int tt = dir ? (Tn-1-s) : s;

    // gate pre-activations from the TDM-staged tile
    v8f acc[4];
    #pragma unroll
    for (int g=0; g<4; ++g){
      #pragma unroll
      for (int r=0;r<8;++r)
        acc[g][r] = stage[(r + hi*8)*G4H + g*Hn + nh + lm];
    }
    __syncthreads();                                    // stage fully consumed

    if (tid < 32 && (s+1) < Tn)                         // prefetch next step's tile
      issue_tdm(dir ? (Tn-2-s) : (s+1));

    #pragma unroll 2
    for (int k = 0; k < Hn; k += 4){
      v2f a;                                            // A frag from h^T (row = batch lm)
      a[0] = hT[(k     + koff)*17 + lm];
      a[1] = hT[(k + 1 + koff)*17 + lm];
      #pragma unroll
      for (int g=0; g<4; ++g){
        const int col = g*Hn + nh + lm;
        v2f bb;
        bb[0] = wlds[(k     + koff)*513 + col];
        bb[1] = wlds[(k + 1 + koff)*513 + col];
        acc[g] = __builtin_amdgcn_wmma_f32_16x16x4_f32(
                   false, a, false, bb, (short)0, acc[g], false, false);
      }
    }

    __syncthreads();                                    // all hT reads done

    float hv[8];
    #pragma unroll
    for (int r=0;r<8;++r){
      const float iv = sigmf(acc[0][r]);
      const float fv = sigmf(acc[1][r]);
      const float gv = tanhf(acc[2][r]);
      const float ov = sigmf(acc[3][r]);
      c[r] = fv*c[r] + iv*gv;
      const float h = ov*tanhf(c[r]);
      hv[r] = h;
      const int b = b0 + r + hi*8;
      Y[(((size_t)b*Tn + tt)*(2*Hn)) + outOff + nh + lm] = h;
    }
    #pragma unroll
    for (int r=0;r<8;++r)
      hT[(nh + lm)*17 + r + hi*8] = hv[r];              // h^T for next step

    if (tid < 32) __builtin_amdgcn_s_wait_tensorcnt(0); // next tile landed
    __syncthreads();
  }
}

// ---------------------------------------------------------------- emissions
__global__ void emissions_k(const float* __restrict__ Yc, const float* __restrict__ Wt,
                            const float* __restrict__ bt_, float* __restrict__ Emis){
  int idx = blockIdx.x*blockDim.x + threadIdx.x;        // < MTOT*8
  int m = idx >> 3, tg = idx & 7;
  const float* y  = &Yc[(size_t)m*(2*Hn)];
  const float* wr = &Wt[(size_t)tg*(2*Hn)];
  float s = bt_[tg];
  #pragma unroll 4
  for (int k=0;k<2*Hn;++k) s += y[k]*wr[k];
  Emis[idx] = s;
}

// ------------------------------------------------ gold emit + trans score
__global__ void score_k(const float* __restrict__ Emis, const int* __restrict__ tags,
                        const float* __restrict__ trans, float* __restrict__ scores){
  __shared__ float red[256];
  const int b = blockIdx.x, tid = threadIdx.x;
  float s = 0.f;
  for (int t = tid; t < Tn; t += 256){
    const int tg = tags[b*Tn + t];
    s += Emis[((size_t)b*Tn + t)*NTAGS + tg];
    if (t > 0) s += trans[tags[b*Tn + t - 1]*NTAGS + tg];
  }
  red[tid] = s; __syncthreads();
  for (int st=128; st>0; st>>=1){ if (tid<st) red[tid]+=red[tid+st]; __syncthreads(); }
  if (tid==0) scores[b] = red[0];
}

// ---------------------------------------------------------- CRF partition
__global__ void partition_k(const float* __restrict__ Emis, const float* __restrict__ trans,
                            float* __restrict__ part){
  const int b = blockIdx.x;
  const int lane = threadIdx.x;                         // 0..31
  const int j = lane & 7;                               // tag (replicated in upper lanes)
  float tr[8];
  #pragma unroll
  for (int i=0;i<8;++i) tr[i] = trans[i*NTAGS + j];
  float fv = Emis[((size_t)b*Tn)*NTAGS + j];
  for (int t = 1; t < Tn; ++t){
    float vals[8];
    #pragma unroll
    for (int i=0;i<8;++i) vals[i] = __shfl(fv, i, 32) + tr[i];
    float m = vals[0];
    #pragma unroll
    for (int i=1;i<8;++i) m = fmaxf(m, vals[i]);
    float ss = 0.f;
    #pragma unroll
    for (int i=0;i<8;++i) ss += __expf(vals[i]-m);
    fv = m + __logf(ss) + Emis[((size_t)b*Tn + t)*NTAGS + j];
  }
  float m = __shfl(fv, 0, 32);
  #pragma unroll
  for (int i=1;i<8;++i) m = fmaxf(m, __shfl(fv, i, 32));
  float ss = 0.f;
  #pragma unroll
  for (int i=0;i<8;++i) ss += __expf(__shfl(fv, i, 32)-m);
  if (lane==0) part[b] = m + __logf(ss);
}

// ---------------------------------------------------------------- final NLL
__global__ void final_k(const float* __restrict__ scores, const float* __restrict__ part,
                        float* __restrict__ out){
  __shared__ float red[64];
  const int tid = threadIdx.x;
  red[tid] = scores[tid] - part[tid];
  __syncthreads();
  for (int st=32; st>0; st>>=1){ if (tid<st) red[tid]+=red[tid+st]; __syncthreads(); }
  if (tid==0) out[0] = -red[0] / (float)Bn;
}

// ---------------------------------------------------------------- launcher
extern "C" void kernel_launch(void* const* d_in, const int* in_sizes, int n_in,
                              void* d_out, int out_size, void* d_ws, size_t ws_size,
                              hipStream_t stream){
  const int*   sent = (const int*)d_in[0];
  const int*   tags = (const int*)d_in[1];
  const float* emb  = (const float*)d_in[2];
  // directions order: l0 fwd, l0 bwd, l1 fwd, l1 bwd
  const float* w_ih[4] = {(const float*)d_in[3],  (const float*)d_in[7],
                          (const float*)d_in[11], (const float*)d_in[15]};
  const float* w_hh[4] = {(const float*)d_in[4],  (const float*)d_in[8],
                          (const float*)d_in[12], (const float*)d_in[16]};
  const float* b_ih[4] = {(const float*)d_in[5],  (const float*)d_in[9],
                          (const float*)d_in[13], (const float*)d_in[17]};
  const float* b_hh[4] = {(const float*)d_in[6],  (const float*)d_in[10],
                          (const float*)d_in[14], (const float*)d_in[18]};
  const float* h2t_w = (const float*)d_in[19];
  const float* h2t_b = (const float*)d_in[20];
  const float* trans = (const float*)d_in[21];

  float* ws     = (float*)d_ws;
  float* x0     = ws;                                   // B*T*128
  float* y      = x0   + (size_t)MTOT*En;               // B*T*256 (reused both layers)
  float* xg_f   = y    + (size_t)MTOT*(2*Hn);           // B*T*512
  float* xg_b   = xg_f + (size_t)MTOT*G4H;              // B*T*512
  float* emis   = xg_b + (size_t)MTOT*G4H;              // B*T*8
  float* scores = emis + (size_t)MTOT*NTAGS;            // B
  float* part   = scores + Bn;                          // B

  embed_k<<<(MTOT*En)/256, 256, 0, stream>>>(sent, emb, x0);

  // layer 0 (K = 128)
  gates_gemm_k<<<MTOT/16, 256, 0, stream>>>(x0, w_ih[0], b_ih[0], b_hh[0], xg_f, En);
  gates_gemm_k<<<MTOT/16, 256, 0, stream>>>(x0, w_ih[1], b_ih[1], b_hh[1], xg_b, En);
  lstm_scan_k<<<dim3(Bn/16, 2), 256, SCAN_LDS_BYTES, stream>>>(xg_f, xg_b, w_hh[0], w_hh[1], y);

  // layer 1 (K = 256); scan output overwrites y after both GEMMs consumed it
  gates_gemm_k<<<MTOT/16, 256, 0, stream>>>(y, w_ih[2], b_ih[2], b_hh[2], xg_f, 2*Hn);
  gates_gemm_k<<<MTOT/16, 256, 0, stream>>>(y, w_ih[3], b_ih[3], b_hh[3], xg_b, 2*Hn);
  lstm_scan_k<<<dim3(Bn/16, 2), 256, SCAN_LDS_BYTES, stream>>>(xg_f, xg_b, w_hh[2], w_hh[3], y);

  emissions_k<<<(MTOT*NTAGS)/256, 256, 0, stream>>>(y, h2t_w, h2t_b, emis);
  score_k<<<Bn, 256, 0, stream>>>(emis, tags, trans, scores);
  partition_k<<<Bn, 32, 0, stream>>>(emis, trans, part);
  final_k<<<1, 64, 0, stream>>>(scores, part, (float*)d_out);
}